// GNNWithAttention_35338990912004
// MI455X (gfx1250) — compile-verified
//
#include <hip/hip_runtime.h>
#include <hip/hip_bf16.h>
#include <math.h>

// ---------------------------------------------------------------------------
// GNN-with-attention (2x GATConv) for gfx1250 (MI455X, wave32, WMMA)
// ---------------------------------------------------------------------------

#define HIDC 128
#define HEADS 4
#define INC 64
#define OUTC 64
#define LN_EPS 1e-5f
#define NEG_SLOPE 0.2f

typedef __attribute__((ext_vector_type(2))) float v2f;
typedef __attribute__((ext_vector_type(8))) float v8f;

// ---------------------------------------------------------------------------
// Weight repack: Wp[k2*COLS + c] = { W[2*k2][c], W[2*k2+1][c] }, k2 in [0,K/2).
// Makes each B fragment (lane<16 -> {W[k][n],W[k+1][n]}, lane>=16 ->
// {W[k+2][n],W[k+3][n]}) a single coalesced global_load_b64.
// ---------------------------------------------------------------------------
__global__ void pack_w_kernel(const float* __restrict__ W,
                              float2* __restrict__ Wp, int K, int COLS) {
  const int i = blockIdx.x * blockDim.x + threadIdx.x;
  const int tot = (K / 2) * COLS;
  if (i >= tot) return;
  const int k2 = i / COLS, c = i % COLS;
  Wp[i] = make_float2(W[(size_t)(2 * k2) * COLS + c],
                      W[(size_t)(2 * k2 + 1) * COLS + c]);
}

// ---------------------------------------------------------------------------
// GEMM: out[M x COLS] = (A[M x K] + abias) @ W[K x COLS] + bias
// One wave = one 16x16 f32 tile via V_WMMA_F32_16X16X4_F32.
// blockDim = 256 (8 waves); block covers ROWS x COLS. A strip staged in LDS
// (abias folded in during staging -> branch-free WMMA loop). LDS rows padded
// by 4 floats so the 16 row-lanes of each ds_load_b64 hit distinct banks.
// Fragment layouts per CDNA5 ISA 7.12.2.
// ---------------------------------------------------------------------------
template <int K, int COLS>
__global__ void gemm_wmma_f32(const float* __restrict__ A,
                              const float* __restrict__ abias,  // len K or null
                              const float2* __restrict__ Wp,    // packed [K/2,COLS]
                              const float* __restrict__ bias,   // len COLS or null
                              float* __restrict__ out, int M) {
  constexpr int colTiles = COLS / 16;
  constexpr int rowStrips = 8 / colTiles;  // 1 (COLS=128) or 2 (COLS=64)
  constexpr int ROWS = rowStrips * 16;
  constexpr int LDK = K + 4;               // padded row stride (floats)
  __shared__ float As[ROWS * LDK];

  const int tid = threadIdx.x;
  const int wave = tid >> 5;
  const int lane = tid & 31;
  const int blockRow0 = blockIdx.x * ROWS;

  // ---- cooperative stage of A (+abias) into LDS ----
  constexpr int NV4 = ROWS * (K / 4);      // float4 slots
  for (int i = tid; i < NV4; i += 256) {
    const int r = i / (K / 4);
    const int kk = (i % (K / 4)) * 4;
    const int grow = blockRow0 + r;
    float4 v = make_float4(0.f, 0.f, 0.f, 0.f);
    if (grow < M) v = *(const float4*)(A + (size_t)grow * K + kk);
    if (abias) {
      const float4 ab = *(const float4*)(abias + kk);
      v.x += ab.x; v.y += ab.y; v.z += ab.z; v.w += ab.w;
    }
    *(float4*)(As + r * LDK + kk) = v;
  }
  __syncthreads();

  const int colTile = wave % colTiles;
  const int rowStrip = wave / colTiles;
  const int row0 = blockRow0 + rowStrip * 16;
  if (row0 >= M) return;  // wave-uniform: EXEC all-ones for WMMA
  const int col0 = colTile * 16;
  const int lmod = lane & 15;
  const int hi = lane >> 4;  // 0: K pair {k,k+1}; 1: {k+2,k+3}

  const float* Alds = As + (rowStrip * 16 + lmod) * LDK;
  const float2* Wl = Wp + col0 + lmod;
  v8f c = {};
#pragma unroll
  for (int k4 = 0; k4 < K / 4; ++k4) {
    const int ka = k4 * 4 + hi * 2;
    v2f a;
    a.x = Alds[ka];
    a.y = Alds[ka + 1];
    const float2 bp = Wl[(size_t)(k4 * 2 + hi) * COLS];
    v2f b;
    b.x = bp.x;
    b.y = bp.y;
    c = __builtin_amdgcn_wmma_f32_16x16x4_f32(false, a, false, b,
                                              (short)0, c, false, false);
  }

  const float bb = bias ? bias[col0 + lmod] : 0.0f;
  const int rbase = row0 + hi * 8;
#pragma unroll
  for (int r = 0; r < 8; ++r) {
    out[(size_t)(rbase + r) * COLS + col0 + lmod] = c[r] + bb;
  }
}

// ---------------------------------------------------------------------------
// Row-wise LayerNorm (+optional pre-bias) + exact GELU. One wave per row of
// 128 channels: lane holds float4, wave32 shfl_xor reduction.
// ---------------------------------------------------------------------------
__device__ __forceinline__ float gelu_exact(float x) {
  return 0.5f * x * (1.0f + erff(x * 0.70710678118654752f));
}

__global__ void ln_gelu_kernel(const float* __restrict__ in,
                               const float* __restrict__ pre_bias,  // 128 or null
                               const float* __restrict__ g,
                               const float* __restrict__ b,
                               float* __restrict__ out, int M) {
  const int wave = threadIdx.x >> 5;
  const int lane = threadIdx.x & 31;
  const int row = blockIdx.x * 8 + wave;
  if (row >= M) return;
  float4 v = *(const float4*)(in + (size_t)row * HIDC + lane * 4);
  if (pre_bias) {
    const float4 pb = *(const float4*)(pre_bias + lane * 4);
    v.x += pb.x; v.y += pb.y; v.z += pb.z; v.w += pb.w;
  }
  float s = v.x + v.y + v.z + v.w;
#pragma unroll
  for (int off = 16; off > 0; off >>= 1) s += __shfl_xor(s, off, 32);
  const float mean = s * (1.0f / HIDC);
  const float dx = v.x - mean, dy = v.y - mean, dz = v.z - mean, dw = v.w - mean;
  float s2 = dx * dx + dy * dy + dz * dz + dw * dw;
#pragma unroll
  for (int off = 16; off > 0; off >>= 1) s2 += __shfl_xor(s2, off, 32);
  const float inv = rsqrtf(s2 * (1.0f / HIDC) + LN_EPS);
  const float4 gg = *(const float4*)(g + lane * 4);
  const float4 bb = *(const float4*)(b + lane * 4);
  float4 o;
  o.x = gelu_exact(dx * inv * gg.x + bb.x);
  o.y = gelu_exact(dy * inv * gg.y + bb.y);
  o.z = gelu_exact(dz * inv * gg.z + bb.z);
  o.w = gelu_exact(dw * inv * gg.w + bb.w);
  *(float4*)(out + (size_t)row * HIDC + lane * 4) = o;
}

// ---------------------------------------------------------------------------
// Per-node attention coefficients: a_src[n,h] = sum_c h[n,h,c]*asrc[h,c]
// One wave per row; 8 lanes per head (32 channels), shfl_xor width 8.
// ---------------------------------------------------------------------------
__global__ void attn_coef_kernel(const float* __restrict__ h,     // [M,128]
                                 const float* __restrict__ asrc,  // [4*32]
                                 const float* __restrict__ adst,  // [4*32]
                                 float* __restrict__ as_out,      // [M,4]
                                 float* __restrict__ ad_out, int M) {
  const int wave = threadIdx.x >> 5;
  const int lane = threadIdx.x & 31;
  const int row = blockIdx.x * 8 + wave;
  if (row >= M) return;
  const float4 v = *(const float4*)(h + (size_t)row * HIDC + lane * 4);
  const float4 a1 = *(const float4*)(asrc + lane * 4);
  const float4 a2 = *(const float4*)(adst + lane * 4);
  float ds = v.x * a1.x + v.y * a1.y + v.z * a1.z + v.w * a1.w;
  float dd = v.x * a2.x + v.y * a2.y + v.z * a2.z + v.w * a2.w;
#pragma unroll
  for (int off = 4; off > 0; off >>= 1) {
    ds += __shfl_xor(ds, off, 8);
    dd += __shfl_xor(dd, off, 8);
  }
  if ((lane & 7) == 0) {
    const int head = lane >> 3;
    as_out[(size_t)row * HEADS + head] = ds;
    ad_out[(size_t)row * HEADS + head] = dd;
  }
}

// ---------------------------------------------------------------------------
// Init per-GAT-layer accumulators: amax=-inf, denom=0, hagg=0.
// ---------------------------------------------------------------------------
__global__ void gat_init_kernel(float* __restrict__ amax,
                                float* __restrict__ denom,
                                float* __restrict__ hagg, int n4, int n128) {
  const int i = blockIdx.x * blockDim.x + threadIdx.x;
  if (i < n4) { amax[i] = -1e30f; denom[i] = 0.0f; }
  if (i < n128) hagg[i] = 0.0f;
}

__device__ __forceinline__ void edge_endpoints(const long long* ei, int E,
                                               int e, int& s, int& d) {
  if (e < E) { s = (int)ei[e]; d = (int)ei[E + e]; }
  else       { s = d = e - E; }  // self loop
}

// Pass 1: per-head segment max of leaky_relu(a_src[src]+a_dst[dst]).
// One thread per edge; 4 heads as float4.
__global__ void edge_alpha_max(const long long* __restrict__ ei, int E, int Nn,
                               const float* __restrict__ as,
                               const float* __restrict__ ad,
                               float* __restrict__ amax) {
  const int e = blockIdx.x * blockDim.x + threadIdx.x;
  if (e >= E + Nn) return;
  int s, d; edge_endpoints(ei, E, e, s, d);
  const float4 a1 = *(const float4*)(as + (size_t)s * HEADS);
  const float4 a2 = *(const float4*)(ad + (size_t)d * HEADS);
  float al[4] = {a1.x + a2.x, a1.y + a2.y, a1.z + a2.z, a1.w + a2.w};
  float* am = amax + (size_t)d * HEADS;
#pragma unroll
  for (int hd = 0; hd < 4; ++hd) {
    const float a = al[hd] >= 0.0f ? al[hd] : NEG_SLOPE * al[hd];
    atomicMax(&am[hd], a);  // global_atomic_max_num_f32
  }
}

// Pass 2: ex = exp(alpha - amax[dst]); store per-edge, accumulate denominator.
__global__ void edge_exp_sum(const long long* __restrict__ ei, int E, int Nn,
                             const float* __restrict__ as,
                             const float* __restrict__ ad,
                             const float* __restrict__ amax,
                             float* __restrict__ denom,
                             float* __restrict__ exb) {
  const int e = blockIdx.x * blockDim.x + threadIdx.x;
  if (e >= E + Nn) return;
  int s, d; edge_endpoints(ei, E, e, s, d);
  const float4 a1 = *(const float4*)(as + (size_t)s * HEADS);
  const float4 a2 = *(const float4*)(ad + (size_t)d * HEADS);
  const float4 am = *(const float4*)(amax + (size_t)d * HEADS);
  float al[4] = {a1.x + a2.x, a1.y + a2.y, a1.z + a2.z, a1.w + a2.w};
  const float amv[4] = {am.x, am.y, am.z, am.w};
  float ex[4];
  float* dn = denom + (size_t)d * HEADS;
#pragma unroll
  for (int hd = 0; hd < 4; ++hd) {
    const float a = al[hd] >= 0.0f ? al[hd] : NEG_SLOPE * al[hd];
    ex[hd] = expf(a - amv[hd]);
    atomicAdd(&dn[hd], ex[hd]);
  }
  *(float4*)(exb + (size_t)e * HEADS) = make_float4(ex[0], ex[1], ex[2], ex[3]);
}

// Pass 3: hagg[dst] += h[src] * w  (one wave per edge; lane owns 4 channels).
__global__ void edge_scatter(const long long* __restrict__ ei, int E, int Nn,
                             const float* __restrict__ h,  // [N,128]
                             const float* __restrict__ exb,
                             const float* __restrict__ denom,
                             float* __restrict__ hagg) {
  const int wave = threadIdx.x >> 5;
  const int lane = threadIdx.x & 31;
  const int e = blockIdx.x * 8 + wave;
  if (e >= E + Nn) return;
  int s, d; edge_endpoints(ei, E, e, s, d);
  const int head = lane >> 3;  // 8 lanes x 4 ch = 32 channels per head
  const float w = exb[(size_t)e * HEADS + head] /
                  (denom[(size_t)d * HEADS + head] + 1e-16f);
  const float4 v = *(const float4*)(h + (size_t)s * HIDC + lane * 4);
  float* o = hagg + (size_t)d * HIDC + lane * 4;
  atomicAdd(o + 0, v.x * w);
  atomicAdd(o + 1, v.y * w);
  atomicAdd(o + 2, v.z * w);
  atomicAdd(o + 3, v.w * w);
}

// ---------------------------------------------------------------------------
// Launcher
// ---------------------------------------------------------------------------
extern "C" void kernel_launch(void* const* d_in, const int* in_sizes, int n_in,
                              void* d_out, int out_size, void* d_ws,
                              size_t ws_size, hipStream_t stream) {
  const float*     x       = (const float*)d_in[0];
  const long long* ei      = (const long long*)d_in[1];
  const float*     W_in    = (const float*)d_in[2];
  const float*     b_in    = (const float*)d_in[3];
  const float*     ln1_g   = (const float*)d_in[4];
  const float*     ln1_b   = (const float*)d_in[5];
  const float*     gat1_W  = (const float*)d_in[6];
  const float*     gat1_as = (const float*)d_in[7];
  const float*     gat1_ad = (const float*)d_in[8];
  const float*     gat1_b  = (const float*)d_in[9];
  const float*     ln2_g   = (const float*)d_in[10];
  const float*     ln2_b   = (const float*)d_in[11];
  const float*     gat2_W  = (const float*)d_in[12];
  const float*     gat2_as = (const float*)d_in[13];
  const float*     gat2_ad = (const float*)d_in[14];
  const float*     gat2_b  = (const float*)d_in[15];
  const float*     W_out   = (const float*)d_in[16];
  const float*     b_out   = (const float*)d_in[17];

  const int Nn = in_sizes[0] / INC;  // 50000
  const int E  = in_sizes[1] / 2;    // 800000
  const int Etot = E + Nn;           // with self loops

  float* ws = (float*)d_ws;
  const size_t N128 = (size_t)Nn * HIDC;
  const size_t N4 = (size_t)Nn * HEADS;
  float*  buf0 = ws;                  // H0 (pre-LN) / hagg1 / hagg2
  float*  buf1 = ws + N128;           // H1 (post-GELU) / H3
  float*  buf2 = ws + 2 * N128;       // Hh (per-layer transformed features)
  float*  asb  = ws + 3 * N128;       // [N,4]
  float*  adb  = asb + N4;            // [N,4]
  float*  amax = adb + N4;            // [N,4]
  float*  den  = amax + N4;           // [N,4]
  float*  exb  = den + N4;            // [E',4]
  float*  pk   = exb + (size_t)Etot * HEADS;
  float2* wpin = (float2*)pk;                       // 64/2 * 128
  float2* wp1  = wpin + (INC / 2) * HIDC;           // 128/2 * 128
  float2* wp2  = wp1 + (HIDC / 2) * HIDC;           // 128/2 * 128
  float2* wpo  = wp2 + (HIDC / 2) * HIDC;           // 128/2 * 64

  const dim3 blk(256);
  const int gRows16 = (Nn + 15) / 16;
  const int gRows32 = (Nn + 31) / 32;
  const int gRowWave = (Nn + 7) / 8;
  const int gInit = ((int)N128 + 255) / 256;
  const int gEdge = (Etot + 255) / 256;
  const int gEdgeW = (Etot + 7) / 8;

  // Pack all weight matrices into B-fragment layout.
  pack_w_kernel<<<((INC / 2) * HIDC + 255) / 256, blk, 0, stream>>>(
      W_in, wpin, INC, HIDC);
  pack_w_kernel<<<((HIDC / 2) * HIDC + 255) / 256, blk, 0, stream>>>(
      gat1_W, wp1, HIDC, HIDC);
  pack_w_kernel<<<((HIDC / 2) * HIDC + 255) / 256, blk, 0, stream>>>(
      gat2_W, wp2, HIDC, HIDC);
  pack_w_kernel<<<((HIDC / 2) * OUTC + 255) / 256, blk, 0, stream>>>(
      W_out, wpo, HIDC, OUTC);

  // 1) H0 = x @ W_in + b_in
  gemm_wmma_f32<INC, HIDC><<<gRows16, blk, 0, stream>>>(
      x, nullptr, wpin, b_in, buf0, Nn);
  // 2) H1 = gelu(ln1(H0))
  ln_gelu_kernel<<<gRowWave, blk, 0, stream>>>(buf0, nullptr, ln1_g, ln1_b,
                                               buf1, Nn);

  // ---- GAT layer 1 ----
  gemm_wmma_f32<HIDC, HIDC><<<gRows16, blk, 0, stream>>>(
      buf1, nullptr, wp1, nullptr, buf2, Nn);
  attn_coef_kernel<<<gRowWave, blk, 0, stream>>>(buf2, gat1_as, gat1_ad,
                                                 asb, adb, Nn);
  gat_init_kernel<<<gInit, blk, 0, stream>>>(amax, den, buf0, (int)N4,
                                             (int)N128);
  edge_alpha_max<<<gEdge, blk, 0, stream>>>(ei, E, Nn, asb, adb, amax);
  edge_exp_sum<<<gEdge, blk, 0, stream>>>(ei, E, Nn, asb, adb, amax, den, exb);
  edge_scatter<<<gEdgeW, blk, 0, stream>>>(ei, E, Nn, buf2, exb, den, buf0);

  // 3) H3 = gelu(ln2(hagg1 + gat1_b))
  ln_gelu_kernel<<<gRowWave, blk, 0, stream>>>(buf0, gat1_b, ln2_g, ln2_b,
                                               buf1, Nn);

  // ---- GAT layer 2 ----
  gemm_wmma_f32<HIDC, HIDC><<<gRows16, blk, 0, stream>>>(
      buf1, nullptr, wp2, nullptr, buf2, Nn);
  attn_coef_kernel<<<gRowWave, blk, 0, stream>>>(buf2, gat2_as, gat2_ad,
                                                 asb, adb, Nn);
  gat_init_kernel<<<gInit, blk, 0, stream>>>(amax, den, buf0, (int)N4,
                                             (int)N128);
  edge_alpha_max<<<gEdge, blk, 0, stream>>>(ei, E, Nn, asb, adb, amax);
  edge_exp_sum<<<gEdge, blk, 0, stream>>>(ei, E, Nn, asb, adb, amax, den, exb);
  edge_scatter<<<gEdgeW, blk, 0, stream>>>(ei, E, Nn, buf2, exb, den, buf0);

  // 4) out = (hagg2 + gat2_b) @ W_out + b_out
  gemm_wmma_f32<HIDC, OUTC><<<gRows32, blk, 0, stream>>>(
      buf0, gat2_b, wpo, b_out, (float*)d_out, Nn);
}